// GraphTransformerEncode_57131654971821
// MI455X (gfx1250) — compile-verified
//
#include <hip/hip_runtime.h>
#include <math.h>

// ---------------------------------------------------------------------------
// GraphTransformerEncode for MI455X (gfx1250, wave32, WMMA)
// D=64, H=4, DEPTH=16, DFF=256. fp32 end-to-end using V_WMMA_F32_16X16X4_F32.
// ---------------------------------------------------------------------------

typedef float v2f __attribute__((ext_vector_type(2)));
typedef float v8f __attribute__((ext_vector_type(8)));

__device__ __forceinline__ v8f wmma_f32_4(v2f a, v2f b, v8f c) {
  // D(16x16,f32) = A(16x4,f32) * B(4x16,f32) + C
  return __builtin_amdgcn_wmma_f32_16x16x4_f32(false, a, false, b, (short)0, c,
                                               false, false);
}

// ---- monotone float<->uint encoding for atomicMax-based segment max --------
__device__ __forceinline__ unsigned enc_f32(float f) {
  unsigned u = __float_as_uint(f);
  return (u & 0x80000000u) ? ~u : (u | 0x80000000u);
}
__device__ __forceinline__ float dec_f32(unsigned u) {
  return (u & 0x80000000u) ? __uint_as_float(u & 0x7fffffffu)
                           : __uint_as_float(~u);
}

// ---------------------------------------------------------------------------
// LayerNorm over 64-float rows: one wave per row, 2 floats/lane, shfl reduce.
// ---------------------------------------------------------------------------
__global__ void ln64_kernel(const float* __restrict__ in,
                            const float* __restrict__ g,
                            const float* __restrict__ b,
                            float* __restrict__ out, int n) {
  int wave = threadIdx.x >> 5, lane = threadIdx.x & 31;
  int row = blockIdx.x * 8 + wave;
  if (row >= n) return;
  float2 val = ((const float2*)(in + (size_t)row * 64))[lane];
  float s = val.x + val.y;
#pragma unroll
  for (int m = 16; m >= 1; m >>= 1) s += __shfl_xor(s, m, 32);
  float mean = s * (1.0f / 64.0f);
  float dx = val.x - mean, dy = val.y - mean;
  float vs = dx * dx + dy * dy;
#pragma unroll
  for (int m = 16; m >= 1; m >>= 1) vs += __shfl_xor(vs, m, 32);
  float rstd = rsqrtf(vs * (1.0f / 64.0f) + 1e-6f);
  float2 gg = ((const float2*)g)[lane];
  float2 bb = ((const float2*)b)[lane];
  float2 o;
  o.x = dx * rstd * gg.x + bb.x;
  o.y = dy * rstd * gg.y + bb.y;
  ((float2*)(out + (size_t)row * 64))[lane] = o;
}

// ---------------------------------------------------------------------------
// One 16-row x@W(64x64)+bias projection computed by a single wave via WMMA.
// A fragment (16x4 f32): lanes 0-15 hold rows, VGPR0/1 = K pair; halves split K.
// ---------------------------------------------------------------------------
__device__ __forceinline__ void proj64(const v2f* afrag, const float* __restrict__ W,
                                       const float* __restrict__ bias,
                                       float* __restrict__ out, int m0, int half,
                                       int mrow) {
#pragma unroll
  for (int nt = 0; nt < 4; ++nt) {
    int col = nt * 16 + mrow;
    float bcol = bias[col];
    v8f acc;
#pragma unroll
    for (int r = 0; r < 8; ++r) acc[r] = bcol;
#pragma unroll
    for (int kk = 0; kk < 16; ++kk) {
      int rw = kk * 4 + 2 * half;
      v2f bf;
      bf.x = W[rw * 64 + col];
      bf.y = W[(rw + 1) * 64 + col];
      acc = wmma_f32_4(afrag[kk], bf, acc);
    }
#pragma unroll
    for (int r = 0; r < 8; ++r)
      out[(size_t)(m0 + half * 8 + r) * 64 + col] = acc[r];
  }
}

__global__ void qkv_kernel(const float* __restrict__ x,
                           const float* __restrict__ Wq, const float* __restrict__ bq,
                           const float* __restrict__ Wk, const float* __restrict__ bk,
                           const float* __restrict__ Wv, const float* __restrict__ bv,
                           float* __restrict__ q, float* __restrict__ k,
                           float* __restrict__ v, int ntiles) {
  int wave = threadIdx.x >> 5;
  int tile = blockIdx.x * 8 + wave;
  if (tile >= ntiles) return;  // wave-uniform: EXEC stays all-ones for WMMA
  int lane = threadIdx.x & 31;
  int half = lane >> 4, mrow = lane & 15;
  int m0 = tile * 16, grow = m0 + mrow;
  v2f afrag[16];
#pragma unroll
  for (int kk = 0; kk < 16; ++kk)
    afrag[kk] = *(const v2f*)(x + (size_t)grow * 64 + kk * 4 + 2 * half);
  proj64(afrag, Wq, bq, q, m0, half, mrow);
  proj64(afrag, Wk, bk, k, m0, half, mrow);
  proj64(afrag, Wv, bv, v, m0, half, mrow);
}

// aggout @ Wd + bd + feature -> out1
__global__ void attnproj_kernel(const float* __restrict__ agg,
                                const float* __restrict__ Wd,
                                const float* __restrict__ bd,
                                const float* __restrict__ feature,
                                float* __restrict__ out1, int ntiles) {
  int wave = threadIdx.x >> 5;
  int tile = blockIdx.x * 8 + wave;
  if (tile >= ntiles) return;
  int lane = threadIdx.x & 31;
  int half = lane >> 4, mrow = lane & 15;
  int m0 = tile * 16, grow = m0 + mrow;
  v2f afrag[16];
#pragma unroll
  for (int kk = 0; kk < 16; ++kk)
    afrag[kk] = *(const v2f*)(agg + (size_t)grow * 64 + kk * 4 + 2 * half);
#pragma unroll
  for (int nt = 0; nt < 4; ++nt) {
    int col = nt * 16 + mrow;
    float bcol = bd[col];
    v8f acc;
#pragma unroll
    for (int r = 0; r < 8; ++r) acc[r] = bcol;
#pragma unroll
    for (int kk = 0; kk < 16; ++kk) {
      int rw = kk * 4 + 2 * half;
      v2f bf;
      bf.x = Wd[rw * 64 + col];
      bf.y = Wd[(rw + 1) * 64 + col];
      acc = wmma_f32_4(afrag[kk], bf, acc);
    }
#pragma unroll
    for (int r = 0; r < 8; ++r) {
      size_t idx = (size_t)(m0 + half * 8 + r) * 64 + col;
      out1[idx] = acc[r] + feature[idx];
    }
  }
}

// ---------------------------------------------------------------------------
// Edge pass 1: logits + atomic segment max (encoded uint) over destination col
// ---------------------------------------------------------------------------
__global__ void edge_logits_kernel(const int* __restrict__ ei,
                                   const float* __restrict__ spv,
                                   const int* __restrict__ erel,
                                   const float* __restrict__ rel_emb,
                                   const float* __restrict__ W_sp,
                                   const float* __restrict__ b_sp,
                                   const float* __restrict__ q,
                                   const float* __restrict__ k,
                                   float* __restrict__ logits,
                                   unsigned* __restrict__ nodemax, int E) {
  int e = blockIdx.x * 256 + threadIdx.x;
  if (e >= E) return;
  int row = ei[e], col = ei[E + e];
  int r = erel[e];
  float sv = spv[e];
  const float4* q4 = (const float4*)q;
  const float4* k4 = (const float4*)k;
  const float4* r4 = (const float4*)rel_emb;
#pragma unroll
  for (int h = 0; h < 4; ++h) {
    float acc = 0.f;
#pragma unroll
    for (int dd = 0; dd < 4; ++dd) {
      float4 rr = r4[r * 16 + h * 4 + dd];
      float4 qe = q4[(size_t)col * 16 + h * 4 + dd];
      float4 ke = k4[(size_t)row * 16 + h * 4 + dd];
      qe.x += rr.x; qe.y += rr.y; qe.z += rr.z; qe.w += rr.w;
      ke.x += rr.x; ke.y += rr.y; ke.z += rr.z; ke.w += rr.w;
      acc += qe.x * ke.x + qe.y * ke.y + qe.z * ke.z + qe.w * ke.w;
    }
    float logit = acc * 0.25f + sv * W_sp[h] + b_sp[h];  // 1/sqrt(16)
    logits[(size_t)e * 4 + h] = logit;
    atomicMax(&nodemax[(size_t)col * 4 + h], enc_f32(logit));
  }
}

// Edge pass 2: ex = exp(logit - max); atomic segment-sum of denominators
__global__ void edge_exp_kernel(const int* __restrict__ ei,
                                float* __restrict__ logits,
                                const unsigned* __restrict__ nodemax,
                                float* __restrict__ den, int E) {
  int e = blockIdx.x * 256 + threadIdx.x;
  if (e >= E) return;
  int col = ei[E + e];
#pragma unroll
  for (int h = 0; h < 4; ++h) {
    float mx = dec_f32(nodemax[(size_t)col * 4 + h]);
    float ex = __expf(logits[(size_t)e * 4 + h] - mx);
    logits[(size_t)e * 4 + h] = ex;
    atomicAdd(&den[(size_t)col * 4 + h], ex);
  }
}

// Edge pass 3: normalize -> attn_w output; scatter-add v[row]*w into agg[col]
__global__ void edge_prop_kernel(const int* __restrict__ ei,
                                 const float* __restrict__ ex,
                                 const float* __restrict__ den,
                                 const float* __restrict__ v,
                                 float* __restrict__ agg,
                                 float* __restrict__ attn_out, int E) {
  int e = blockIdx.x * 256 + threadIdx.x;
  if (e >= E) return;
  int row = ei[e], col = ei[E + e];
  float w[4];
#pragma unroll
  for (int h = 0; h < 4; ++h) {
    w[h] = ex[(size_t)e * 4 + h] / (den[(size_t)col * 4 + h] + 1e-16f);
    attn_out[(size_t)e * 4 + h] = w[h];
  }
  const float4* v4 = (const float4*)v;
#pragma unroll
  for (int i = 0; i < 16; ++i) {
    float4 mv = v4[(size_t)row * 16 + i];
    float wh = w[i >> 2];
    float* dst = agg + (size_t)col * 64 + i * 4;
    atomicAdd(dst + 0, mv.x * wh);
    atomicAdd(dst + 1, mv.y * wh);
    atomicAdd(dst + 2, mv.z * wh);
    atomicAdd(dst + 3, mv.w * wh);
  }
}

// ---------------------------------------------------------------------------
// Fused FFN: relu(h@W1+b1)@W2+b2 + out1, staged through LDS (no 100MB temp).
// 2 waves/block, each wave owns a 16-row tile and a padded 16x260 LDS panel.
// ---------------------------------------------------------------------------
__global__ __launch_bounds__(64) void ffn_kernel(
    const float* __restrict__ h, const float* __restrict__ W1,
    const float* __restrict__ b1, const float* __restrict__ W2,
    const float* __restrict__ b2, const float* __restrict__ out1,
    float* __restrict__ out2, int ntiles) {
  __shared__ float lds[2][16][260];  // stride 260: rows hit distinct banks
  int wave = threadIdx.x >> 5;
  int tile = blockIdx.x * 2 + wave;
  if (tile >= ntiles) tile = ntiles - 1;  // clamp (no divergent barrier)
  int lane = threadIdx.x & 31;
  int half = lane >> 4, mrow = lane & 15;
  int m0 = tile * 16, grow = m0 + mrow;
  v2f afrag[16];
#pragma unroll
  for (int kk = 0; kk < 16; ++kk)
    afrag[kk] = *(const v2f*)(h + (size_t)grow * 64 + kk * 4 + 2 * half);
  // ---- stage 1: u = relu(h@W1 + b1), 16x256 tile into LDS ----
#pragma unroll
  for (int nt = 0; nt < 16; ++nt) {
    int col = nt * 16 + mrow;
    float bcol = b1[col];
    v8f acc;
#pragma unroll
    for (int r = 0; r < 8; ++r) acc[r] = bcol;
#pragma unroll
    for (int kk = 0; kk < 16; ++kk) {
      int rw = kk * 4 + 2 * half;
      v2f bf;
      bf.x = W1[rw * 256 + col];
      bf.y = W1[(rw + 1) * 256 + col];
      acc = wmma_f32_4(afrag[kk], bf, acc);
    }
#pragma unroll
    for (int r = 0; r < 8; ++r)
      lds[wave][half * 8 + r][col] = fmaxf(acc[r], 0.0f);
  }
  __syncthreads();
  // ---- stage 2: u(16x256) @ W2(256x64) + b2 + out1 -> out2 ----
#pragma unroll
  for (int nt = 0; nt < 4; ++nt) {
    int col = nt * 16 + mrow;
    float bcol = b2[col];
    v8f acc;
#pragma unroll
    for (int r = 0; r < 8; ++r) acc[r] = bcol;
#pragma unroll
    for (int kk = 0; kk < 64; ++kk) {
      v2f af;
      af.x = lds[wave][mrow][kk * 4 + 2 * half];
      af.y = lds[wave][mrow][kk * 4 + 2 * half + 1];
      int rw = kk * 4 + 2 * half;
      v2f bf;
      bf.x = W2[rw * 64 + col];
      bf.y = W2[(rw + 1) * 64 + col];
      acc = wmma_f32_4(af, bf, acc);
    }
#pragma unroll
    for (int r = 0; r < 8; ++r) {
      size_t idx = (size_t)(m0 + half * 8 + r) * 64 + col;
      out2[idx] = acc[r] + out1[idx];
    }
  }
}

// ---------------------------------------------------------------------------
extern "C" void kernel_launch(void* const* d_in, const int* in_sizes, int n_in,
                              void* d_out, int out_size, void* d_ws,
                              size_t ws_size, hipStream_t stream) {
  const float* feature = (const float*)d_in[0];
  const int* ei        = (const int*)d_in[1];
  const float* spv     = (const float*)d_in[2];
  const int* erel      = (const int*)d_in[3];
  const float* rel_emb = (const float*)d_in[4];
  const float* W_sp    = (const float*)d_in[5];
  const float* b_sp    = (const float*)d_in[6];
  const float* Wq = (const float*)d_in[7],  *bq = (const float*)d_in[8];
  const float* Wk = (const float*)d_in[9],  *bk = (const float*)d_in[10];
  const float* Wv = (const float*)d_in[11], *bv = (const float*)d_in[12];
  const float* Wd = (const float*)d_in[13], *bd = (const float*)d_in[14];
  const float* W1 = (const float*)d_in[15], *b1 = (const float*)d_in[16];
  const float* W2 = (const float*)d_in[17], *b2 = (const float*)d_in[18];
  const float* g1 = (const float*)d_in[19], *be1 = (const float*)d_in[20];
  const float* g2 = (const float*)d_in[21], *be2 = (const float*)d_in[22];

  const int n = in_sizes[0] / 64;
  const int E = in_sizes[2];
  const int ntiles = (n + 15) / 16;

  // workspace layout (floats), reused across phases
  float* ws = (float*)d_ws;
  float* x      = ws;                      // [n,64] LN1 output; reused as h2
  float* q      = x + (size_t)n * 64;      // [n,64]; reused as aggout
  float* k      = q + (size_t)n * 64;      // [n,64]; reused as out1
  float* v      = k + (size_t)n * 64;      // [n,64]
  float* logits = v + (size_t)n * 64;      // [E,4]; reused as ex
  unsigned* nodemax = (unsigned*)(logits + (size_t)E * 4);  // [n,4]
  float* den    = (float*)(nodemax + (size_t)n * 4);        // [n,4]

  float* out2_dst = (float*)d_out;                 // [n,64]
  float* attn_dst = (float*)d_out + (size_t)n * 64; // [E,4]

  // 1) LN1
  ln64_kernel<<<(n + 7) / 8, 256, 0, stream>>>(feature, g1, be1, x, n);
  // 2) q,k,v projections (WMMA fp32)
  qkv_kernel<<<(ntiles + 7) / 8, 256, 0, stream>>>(x, Wq, bq, Wk, bk, Wv, bv,
                                                   q, k, v, ntiles);
  // 3) segment softmax scratch init
  hipMemsetAsync(nodemax, 0, (size_t)n * 4 * sizeof(unsigned), stream);
  hipMemsetAsync(den, 0, (size_t)n * 4 * sizeof(float), stream);
  // 4) edge logits + segment max
  edge_logits_kernel<<<(E + 255) / 256, 256, 0, stream>>>(
      ei, spv, erel, rel_emb, W_sp, b_sp, q, k, logits, nodemax, E);
  // 5) exp + segment sum
  edge_exp_kernel<<<(E + 255) / 256, 256, 0, stream>>>(ei, logits, nodemax,
                                                       den, E);
  // 6) normalize + scatter-add propagate (q buffer becomes aggout)
  float* aggout = q;
  hipMemsetAsync(aggout, 0, (size_t)n * 64 * sizeof(float), stream);
  edge_prop_kernel<<<(E + 255) / 256, 256, 0, stream>>>(ei, logits, den, v,
                                                        aggout, attn_dst, E);
  // 7) output projection + residual (k buffer becomes out1)
  float* out1 = k;
  attnproj_kernel<<<(ntiles + 7) / 8, 256, 0, stream>>>(aggout, Wd, bd,
                                                        feature, out1, ntiles);
  // 8) LN2 (x buffer becomes h2)
  float* h2 = x;
  ln64_kernel<<<(n + 7) / 8, 256, 0, stream>>>(out1, g2, be2, h2, n);
  // 9) fused FFN + residual -> out2
  ffn_kernel<<<(ntiles + 1) / 2, 64, 0, stream>>>(h2, W1, b1, W2, b2, out1,
                                                  out2_dst, ntiles);
}